// MambaDecoderBlock_2499670966924
// MI455X (gfx1250) — compile-verified
//
#include <hip/hip_runtime.h>
#include <math.h>

// ---------------------------------------------------------------------------
// Mamba decoder block for gfx1250 (MI455X), fp32, using V_WMMA_F32_16X16X4_F32
// ---------------------------------------------------------------------------

#define BATCH   2
#define SEQ     2048
#define DMODEL  512
#define DINNER  1024
#define DSTATE  16
#define DTRANK  32
#define DCONV   4
#define MROWS   (BATCH * SEQ)        // 4096

typedef float v2f __attribute__((ext_vector_type(2)));
typedef float v8f __attribute__((ext_vector_type(8)));

__device__ __forceinline__ float softplus_f(float v) {
    // jax.nn.softplus = log1p(exp(x)), stable form
    return (v > 20.0f) ? v : log1pf(expf(v));
}

// ---------------------------------------------------------------------------
// Generic row-major fp32 WMMA GEMM: C[M,N] = A[M,K] * B[K,N]  (+ epilogue)
// Block = 128 threads (4 waves). Each wave computes a 32(M) x 64(N) tile
// (2 A fragments x 4 B fragments = 8 accumulators), so one block covers
// 128x64. Grid = (M/128, N/64). K must be a multiple of 4.
// EPI == 1: v = softplus(v + bias[col])
// ---------------------------------------------------------------------------
template <int EPI>
__global__ __launch_bounds__(128) void wmma_gemm_f32(
    const float* __restrict__ A, const float* __restrict__ B,
    float* __restrict__ C, const float* __restrict__ bias,
    int K, int lda, int ldb, int ldc)
{
    const int wave   = threadIdx.x >> 5;
    const int lane   = threadIdx.x & 31;
    const int half   = lane >> 4;      // 0: lanes 0-15, 1: lanes 16-31
    const int lane16 = lane & 15;

    const int tileM = (blockIdx.x * 4 + wave) * 32;
    const int tileN = blockIdx.y * 64;

    v8f acc00 = {}; v8f acc01 = {}; v8f acc02 = {}; v8f acc03 = {};
    v8f acc10 = {}; v8f acc11 = {}; v8f acc12 = {}; v8f acc13 = {};

    const float* Arow0 = A + (size_t)(tileM + lane16) * lda;        // rows 0..15
    const float* Arow1 = A + (size_t)(tileM + 16 + lane16) * lda;   // rows 16..31

    #pragma unroll 4
    for (int k = 0; k < K; k += 4) {
        // A fragments: 16x4; lanes 0-15 hold K={k,k+1}, lanes 16-31 K={k+2,k+3}
        v2f a0, a1;
        a0.x = Arow0[k + 2 * half + 0];
        a0.y = Arow0[k + 2 * half + 1];
        a1.x = Arow1[k + 2 * half + 0];
        a1.y = Arow1[k + 2 * half + 1];

        // B fragments: 4x16 per WMMA; row (k+2*half+j) striped across lanes
        const float* B0 = B + (size_t)(k + 2 * half) * ldb + tileN + lane16;
        const float* B1 = B0 + ldb;

        v2f b0; b0.x = B0[0];  b0.y = B1[0];
        v2f b1; b1.x = B0[16]; b1.y = B1[16];
        v2f b2; b2.x = B0[32]; b2.y = B1[32];
        v2f b3; b3.x = B0[48]; b3.y = B1[48];

        acc00 = __builtin_amdgcn_wmma_f32_16x16x4_f32(false, a0, false, b0, (short)0, acc00, false, false);
        acc01 = __builtin_amdgcn_wmma_f32_16x16x4_f32(false, a0, false, b1, (short)0, acc01, false, false);
        acc02 = __builtin_amdgcn_wmma_f32_16x16x4_f32(false, a0, false, b2, (short)0, acc02, false, false);
        acc03 = __builtin_amdgcn_wmma_f32_16x16x4_f32(false, a0, false, b3, (short)0, acc03, false, false);
        acc10 = __builtin_amdgcn_wmma_f32_16x16x4_f32(false, a1, false, b0, (short)0, acc10, false, false);
        acc11 = __builtin_amdgcn_wmma_f32_16x16x4_f32(false, a1, false, b1, (short)0, acc11, false, false);
        acc12 = __builtin_amdgcn_wmma_f32_16x16x4_f32(false, a1, false, b2, (short)0, acc12, false, false);
        acc13 = __builtin_amdgcn_wmma_f32_16x16x4_f32(false, a1, false, b3, (short)0, acc13, false, false);
    }

    // Store: C/D layout — VGPR r, lanes 0-15: M=r, lanes 16-31: M=r+8
    #pragma unroll
    for (int mi = 0; mi < 2; ++mi) {
        const v8f* accs[4] = {
            mi == 0 ? &acc00 : &acc10, mi == 0 ? &acc01 : &acc11,
            mi == 0 ? &acc02 : &acc12, mi == 0 ? &acc03 : &acc13 };
        #pragma unroll
        for (int r = 0; r < 8; ++r) {
            const int row = tileM + mi * 16 + r + 8 * half;
            float* Crow = C + (size_t)row * ldc + tileN + lane16;
            float v0 = (*accs[0])[r], v1 = (*accs[1])[r];
            float v2 = (*accs[2])[r], v3 = (*accs[3])[r];
            if (EPI == 1) {
                v0 = softplus_f(v0 + bias[tileN + lane16 +  0]);
                v1 = softplus_f(v1 + bias[tileN + lane16 + 16]);
                v2 = softplus_f(v2 + bias[tileN + lane16 + 32]);
                v3 = softplus_f(v3 + bias[tileN + lane16 + 48]);
            }
            Crow[0]  = v0;
            Crow[16] = v1;
            Crow[32] = v2;
            Crow[48] = v3;
        }
    }
}

// ---------------------------------------------------------------------------
// Depthwise causal conv (width 4) + bias + SiLU over the xi half of xz.
// xz is (MROWS, 2*DINNER); xi = xz[:, :DINNER]. Output xc is (MROWS, DINNER).
// ---------------------------------------------------------------------------
__global__ void conv_silu_kernel(const float* __restrict__ xz,
                                 const float* __restrict__ conv_w,
                                 const float* __restrict__ conv_b,
                                 float* __restrict__ xc)
{
    int idx = blockIdx.x * blockDim.x + threadIdx.x;
    if (idx >= MROWS * DINNER) return;
    const int d   = idx & (DINNER - 1);
    const int row = idx >> 10;          // b*SEQ + l
    const int l   = row & (SEQ - 1);

    float acc = conv_b[d];
    #pragma unroll
    for (int j = 0; j < DCONV; ++j) {
        const int ls = l - (DCONV - 1) + j;
        if (ls >= 0)
            acc += conv_w[d * DCONV + j] *
                   xz[(size_t)(row - (DCONV - 1) + j) * (2 * DINNER) + d];
    }
    xc[(size_t)row * DINNER + d] = acc / (1.0f + expf(-acc));
}

// ---------------------------------------------------------------------------
// Selective scan. One lane per (b, d, s). 16 consecutive lanes share one d;
// y_t = sum_s h*C reduced with shfl_xor across the 16-lane group.
// ---------------------------------------------------------------------------
__global__ void scan_kernel(const float* __restrict__ dt,
                            const float* __restrict__ xc,
                            const float* __restrict__ dbl,  // (MROWS, 64): [dt_r|B|C]
                            const float* __restrict__ A_log,
                            float* __restrict__ y)
{
    const int t = blockIdx.x * blockDim.x + threadIdx.x;   // 32768 total
    const int s = t & (DSTATE - 1);
    const int d = (t >> 4) & (DINNER - 1);
    const int b = t >> 14;

    const float A = -expf(A_log[d * DSTATE + s]);
    float h = 0.0f;

    for (int l = 0; l < SEQ; ++l) {
        const int row = b * SEQ + l;
        const float dtv = dt[(size_t)row * DINNER + d];
        const float xv  = xc[(size_t)row * DINNER + d];
        const float Bv  = dbl[(size_t)row * 64 + DTRANK + s];
        const float Cv  = dbl[(size_t)row * 64 + DTRANK + DSTATE + s];

        h = h * expf(dtv * A) + (dtv * xv) * Bv;

        float c = h * Cv;
        #pragma unroll
        for (int m = 1; m < DSTATE; m <<= 1)
            c += __shfl_xor(c, m, 32);
        if (s == 0)
            y[(size_t)row * DINNER + d] = c;
    }
}

// ---------------------------------------------------------------------------
// Gate: y = (y + xc * D) * silu(z); z = xz[:, DINNER + d]. In-place on y.
// ---------------------------------------------------------------------------
__global__ void gate_kernel(float* __restrict__ y,
                            const float* __restrict__ xc,
                            const float* __restrict__ xz,
                            const float* __restrict__ Dvec)
{
    int idx = blockIdx.x * blockDim.x + threadIdx.x;
    if (idx >= MROWS * DINNER) return;
    const int d   = idx & (DINNER - 1);
    const int row = idx >> 10;
    const float z = xz[(size_t)row * (2 * DINNER) + DINNER + d];
    y[idx] = (y[idx] + xc[idx] * Dvec[d]) * (z / (1.0f + expf(-z)));
}

// ---------------------------------------------------------------------------
// Residual + LayerNorm. One wave per row of 512; 16 elements per lane.
// ---------------------------------------------------------------------------
__global__ void ln_kernel(const float* __restrict__ proj,
                          const float* __restrict__ x,
                          const float* __restrict__ gamma,
                          const float* __restrict__ beta,
                          float* __restrict__ out)
{
    const int row  = blockIdx.x;
    const int lane = threadIdx.x;          // 0..31
    float vals[DMODEL / 32];

    float sum = 0.0f;
    #pragma unroll
    for (int i = 0; i < DMODEL / 32; ++i) {
        const int col = lane + i * 32;
        const float v = proj[(size_t)row * DMODEL + col] +
                        x[(size_t)row * DMODEL + col];
        vals[i] = v;
        sum += v;
    }
    #pragma unroll
    for (int m = 1; m < 32; m <<= 1) sum += __shfl_xor(sum, m, 32);
    const float mu = sum * (1.0f / DMODEL);

    float var = 0.0f;
    #pragma unroll
    for (int i = 0; i < DMODEL / 32; ++i) {
        const float dl = vals[i] - mu;
        var += dl * dl;
    }
    #pragma unroll
    for (int m = 1; m < 32; m <<= 1) var += __shfl_xor(var, m, 32);
    var *= (1.0f / DMODEL);
    const float inv = rsqrtf(var + 1e-5f);

    #pragma unroll
    for (int i = 0; i < DMODEL / 32; ++i) {
        const int col = lane + i * 32;
        out[(size_t)row * DMODEL + col] =
            (vals[i] - mu) * inv * gamma[col] + beta[col];
    }
}

// ---------------------------------------------------------------------------
// Host-side orchestration
// ---------------------------------------------------------------------------
extern "C" void kernel_launch(void* const* d_in, const int* in_sizes, int n_in,
                              void* d_out, int out_size, void* d_ws, size_t ws_size,
                              hipStream_t stream)
{
    (void)in_sizes; (void)n_in; (void)out_size; (void)ws_size;

    const float* x       = (const float*)d_in[0];   // (2,2048,512)
    const float* W_in    = (const float*)d_in[1];   // (512, 2048)
    const float* conv_w  = (const float*)d_in[2];   // (1024, 4)
    const float* conv_b  = (const float*)d_in[3];   // (1024,)
    const float* W_xproj = (const float*)d_in[4];   // (1024, 64)
    const float* W_dt    = (const float*)d_in[5];   // (32, 1024)
    const float* b_dt    = (const float*)d_in[6];   // (1024,)
    const float* A_log   = (const float*)d_in[7];   // (1024, 16)
    const float* Dvec    = (const float*)d_in[8];   // (1024,)
    const float* W_out   = (const float*)d_in[9];   // (1024, 512)
    const float* gamma   = (const float*)d_in[10];  // (512,)
    const float* beta    = (const float*)d_in[11];  // (512,)
    float* out = (float*)d_out;

    char* ws = (char*)d_ws;
    float* xz   = (float*)(ws + 0);                          // 4096x2048 (32 MB)
    float* xc   = (float*)(ws + (size_t)33554432);           // 4096x1024 (16 MB)
    float* dbl  = (float*)(ws + (size_t)50331648);           // 4096x64   ( 1 MB)
    float* dtb  = (float*)(ws + (size_t)51380224);           // 4096x1024 (16 MB)
    float* ybuf = (float*)(ws + (size_t)68157440);           // 4096x1024 (16 MB)
    float* proj = xz;  // GEMM-out output aliases xz (xz dead by then)

    const int EW = MROWS * DINNER;                           // 4 Mi elements

    // 1) xz = x @ W_in            (M=4096, N=2048, K=512)
    wmma_gemm_f32<0><<<dim3(MROWS / 128, 2 * DINNER / 64), 128, 0, stream>>>(
        x, W_in, xz, nullptr, DMODEL, DMODEL, 2 * DINNER, 2 * DINNER);

    // 2) xc = silu(causal_conv(xi) + conv_b)
    conv_silu_kernel<<<(EW + 255) / 256, 256, 0, stream>>>(xz, conv_w, conv_b, xc);

    // 3) dbl = xc @ W_xproj       (M=4096, N=64, K=1024)
    wmma_gemm_f32<0><<<dim3(MROWS / 128, 1), 128, 0, stream>>>(
        xc, W_xproj, dbl, nullptr, DINNER, DINNER, 64, 64);

    // 4) dt = softplus(dtr @ W_dt + b_dt)   (A = dbl[:, :32], lda=64, K=32)
    wmma_gemm_f32<1><<<dim3(MROWS / 128, DINNER / 64), 128, 0, stream>>>(
        dbl, W_dt, dtb, b_dt, DTRANK, 64, DINNER, DINNER);

    // 5) selective scan -> ybuf
    scan_kernel<<<(BATCH * DINNER * DSTATE) / 256, 256, 0, stream>>>(
        dtb, xc, dbl, A_log, ybuf);

    // 6) gate (in place on ybuf)
    gate_kernel<<<(EW + 255) / 256, 256, 0, stream>>>(ybuf, xc, xz, Dvec);

    // 7) proj = ybuf @ W_out      (M=4096, N=512, K=1024)
    wmma_gemm_f32<0><<<dim3(MROWS / 128, DMODEL / 64), 128, 0, stream>>>(
        ybuf, W_out, proj, nullptr, DINNER, DINNER, DMODEL, DMODEL);

    // 8) residual + layernorm -> out
    ln_kernel<<<MROWS, 32, 0, stream>>>(proj, x, gamma, beta, out);
}